// CBFocalSegmenterV0_34471407518074
// MI455X (gfx1250) — compile-verified
//
#include <hip/hip_runtime.h>
#include <math.h>

#ifndef __has_builtin
#define __has_builtin(x) 0
#endif

// Problem dimensions (fixed by the reference).
#define CB_B   8
#define CB_C   21
#define CB_HW  (512 * 512)          // 262144
#define CB_N   (CB_B * CB_HW)       // 2097152 pixels

// Tiling for the TDM path.
#define TILE_P        64                      // pixels per tile (per wave)
#define TILE_FLOATS   (CB_C * TILE_P)         // 1344 floats = 5376 B per tensor tile
#define WAVES_PER_BLK 2
#define BLK_THREADS   (WAVES_PER_BLK * 32)    // 64
#define NBLK          4096
#define TILES_TOTAL   (CB_N / TILE_P)         // 32768 (divides exactly)

// Device pass picks the CDNA5 Tensor Data Mover path when the builtin exists.
// This toolchain (clang-23 / therock-10.0) uses the 6-argument form:
//   (u32x4 g0, i32x8 g1, i32x4 g2, i32x4 g3, i32x8 extra, i32 cpol)
#if defined(__HIP_DEVICE_COMPILE__) && \
    __has_builtin(__builtin_amdgcn_tensor_load_to_lds) && \
    __has_builtin(__builtin_amdgcn_s_wait_tensorcnt)
#define USE_TDM 1
#else
#define USE_TDM 0
#endif

typedef unsigned int cb_u32x4 __attribute__((ext_vector_type(4)));
typedef int          cb_i32x8 __attribute__((ext_vector_type(8)));
typedef int          cb_i32x4 __attribute__((ext_vector_type(4)));

#if USE_TDM
// Issue one TDM load of a 2D tile: 21 rows (channels) x TILE_P floats,
// row stride = CB_HW floats, packed contiguously into LDS at ldsOff.
__device__ __forceinline__ void cb_tdm_issue(const float* gptr, unsigned ldsOff) {
  unsigned long long ga = (unsigned long long)(const void*)gptr;

  cb_u32x4 g0;
  g0.x = 1u;                                              // count=1 (valid descriptor)
  g0.y = ldsOff;                                          // LDS byte address
  g0.z = (unsigned)ga;                                    // global_addr[31:0]
  g0.w = (unsigned)((ga >> 32) & 0x01FFFFFFull)           // global_addr[56:32]
       | (2u << 30);                                      // type=2 ("image")

  cb_i32x8 g1;
  g1[0] = (int)(2u << 16);                                // data_size=2 -> 4 bytes
  g1[1] = (int)(((unsigned)CB_HW & 0xFFFFu) << 16);       // tensor_dim0[15:0] @ bits63:48
  g1[2] = (int)((((unsigned)CB_HW >> 16) & 0xFFFFu)       // tensor_dim0[31:16]
       | ((unsigned)CB_C << 16));                         // tensor_dim1[15:0] = 21
  g1[3] = (int)((unsigned)TILE_P << 16);                  // tensor_dim1[31:16]=0, tile_dim0=64
  g1[4] = (int)(unsigned)CB_C;                            // tile_dim1=21, tile_dim2=0
  g1[5] = (int)(unsigned)CB_HW;                           // tensor_dim0_stride[31:0]
  g1[6] = 0;                                              // stride[47:32]=0, dim1_stride[15:0]=0
  g1[7] = 0;                                              // dim1_stride[47:16]=0

  cb_i32x4 g2; g2[0] = 1; g2[1] = 1; g2[2] = 0; g2[3] = 0;        // tensor_dim2/3 = 1 (unused)
  cb_i32x4 g3; g3[0] = 0; g3[1] = (int)(1u << 16); g3[2] = 0; g3[3] = 0; // tensor_dim4 = 1

  cb_i32x8 gx;                                            // extra group (zero-filled)
  gx[0] = 0; gx[1] = 0; gx[2] = 0; gx[3] = 0;
  gx[4] = 0; gx[5] = 0; gx[6] = 0; gx[7] = 0;

  __builtin_amdgcn_tensor_load_to_lds(g0, g1, g2, g3, gx, 0);
}
#endif

// Per-pixel focal math + scatter into the block's 21-bin LDS accumulators.
__device__ __forceinline__ void cb_pixel(const float* x, float m, float xl, int labl,
                                         float* s_sum, float* s_cnt) {
  float sse = 0.0f;
#pragma unroll
  for (int c = 0; c < CB_C; ++c) sse += __expf(x[c] - m);
  float lse = m + __logf(sse);
  float lj  = xl - lse;              // log p_label
  float pj  = __expf(lj);
  float om  = 1.0f - pj;
  float rv  = om * om * lj;          // (1-p)^2 * log p   (gamma = 2)
  atomicAdd(&s_sum[labl], rv);
  atomicAdd(&s_cnt[labl], 1.0f);
}

__global__ void cb_zero(float* acc) {
  if (threadIdx.x < 2 * CB_C) acc[threadIdx.x] = 0.0f;
}

__global__ __launch_bounds__(BLK_THREADS)
void cb_main(const float* __restrict__ pred, const float* __restrict__ tgt,
             float* __restrict__ acc) {
  __shared__ float s_sum[CB_C];
  __shared__ float s_cnt[CB_C];
  if (threadIdx.x < CB_C) { s_sum[threadIdx.x] = 0.0f; s_cnt[threadIdx.x] = 0.0f; }
  __syncthreads();

#if USE_TDM
  // [wave][buffer][pred|tgt][21*TILE_P] — 43008 B, double-buffered per wave.
  __shared__ float s_tiles[WAVES_PER_BLK][2][2][TILE_FLOATS];

  const int wave   = __builtin_amdgcn_readfirstlane((int)(threadIdx.x >> 5));
  const int lane   = (int)(threadIdx.x & 31);
  const int gwave  = (int)blockIdx.x * WAVES_PER_BLK + wave;
  const int nwaves = (int)gridDim.x * WAVES_PER_BLK;
  const int tilesPerBatch = CB_HW / TILE_P;

  int T = 0;
  if (gwave < TILES_TOTAL) T = (TILES_TOTAL - gwave + nwaves - 1) / nwaves;

  auto tileBase = [&](int t, const float* basep) -> const float* {
    int b  = t / tilesPerBatch;
    int r0 = (t - b * tilesPerBatch) * TILE_P;
    return basep + (size_t)b * CB_C * CB_HW + r0;
  };
  auto ldsOff = [&](int buf, int which) -> unsigned {
    // Generic pointers to LDS carry the byte offset in the low 32 bits.
    return (unsigned)(unsigned long long)(const void*)&s_tiles[wave][buf][which][0];
  };

  if (T > 0) {
    int t0 = gwave;
    cb_tdm_issue(tileBase(t0, pred), ldsOff(0, 0));
    cb_tdm_issue(tileBase(t0, tgt ), ldsOff(0, 1));
  }
  if (T > 1) {
    int t1 = gwave + nwaves;
    cb_tdm_issue(tileBase(t1, pred), ldsOff(1, 0));
    cb_tdm_issue(tileBase(t1, tgt ), ldsOff(1, 1));
  }

  for (int k = 0; k < T; ++k) {
    if (k + 1 < T) __builtin_amdgcn_s_wait_tensorcnt(2);  // current tile's 2 DMAs done
    else           __builtin_amdgcn_s_wait_tensorcnt(0);
    __asm__ volatile("" ::: "memory");                    // keep LDS reads below the wait

    const int buf = k & 1;
    const float* pb = &s_tiles[wave][buf][0][0];
    const float* tb = &s_tiles[wave][buf][1][0];

#pragma unroll
    for (int kk = 0; kk < TILE_P / 32; ++kk) {
      const int pix = lane + 32 * kk;
      float x[CB_C];
      float m = -3.402823466e38f, xl = 0.0f;
      int   labl = 0;
#pragma unroll
      for (int c = 0; c < CB_C; ++c) {
        float xv = pb[c * TILE_P + pix];
        float tv = tb[c * TILE_P + pix];
        x[c] = xv;
        m = fmaxf(m, xv);
        bool hit = (tv == 1.0f);      // one-hot target selects the label
        xl   = hit ? xv : xl;
        labl = hit ? c  : labl;
      }
      cb_pixel(x, m, xl, labl, s_sum, s_cnt);
    }

    // Ensure our LDS reads/atomics retired before the TDM overwrites this buffer.
    __asm__ volatile("s_wait_dscnt 0" ::: "memory");
    if (k + 2 < T) {
      int tn = gwave + (k + 2) * nwaves;
      cb_tdm_issue(tileBase(tn, pred), ldsOff(buf, 0));
      cb_tdm_issue(tileBase(tn, tgt ), ldsOff(buf, 1));
    }
  }
#else
  // Portable fallback: direct strided streaming loads (coalesced across lanes).
  const int tid = (int)(blockIdx.x * blockDim.x + threadIdx.x);
  const int nth = (int)(gridDim.x * blockDim.x);
  for (int i = tid; i < CB_N; i += nth) {
    int b = i / CB_HW;
    int r = i - b * CB_HW;
    const float* pp = pred + (size_t)b * CB_C * CB_HW + r;
    const float* tp = tgt  + (size_t)b * CB_C * CB_HW + r;
    float x[CB_C];
    float m = -3.402823466e38f, xl = 0.0f;
    int   labl = 0;
#pragma unroll
    for (int c = 0; c < CB_C; ++c) {
      float xv = __builtin_nontemporal_load(pp + (size_t)c * CB_HW);
      float tv = __builtin_nontemporal_load(tp + (size_t)c * CB_HW);
      x[c] = xv;
      m = fmaxf(m, xv);
      bool hit = (tv == 1.0f);
      xl   = hit ? xv : xl;
      labl = hit ? c  : labl;
    }
    cb_pixel(x, m, xl, labl, s_sum, s_cnt);
  }
#endif

  __syncthreads();
  if (threadIdx.x < CB_C) {
    atomicAdd(&acc[threadIdx.x],        s_sum[threadIdx.x]);
    atomicAdd(&acc[CB_C + threadIdx.x], s_cnt[threadIdx.x]);
  }
}

__global__ void cb_final(const float* __restrict__ acc, float* __restrict__ out) {
  if (threadIdx.x == 0) {
    const float nf   = (float)CB_N;
    const float beta = (nf - 1.0f) / nf;   // exactly 1 - 2^-21 in f32
    float total = 0.0f;
    for (int c = 0; c < CB_C; ++c) {
      float nc = acc[CB_C + c];
      float w  = (1.0f - beta) / (1.0f - powf(beta, nc) + 1e-6f);
      total += w * acc[c];
    }
    out[0] = -total / nf;
  }
}

extern "C" void kernel_launch(void* const* d_in, const int* in_sizes, int n_in,
                              void* d_out, int out_size, void* d_ws, size_t ws_size,
                              hipStream_t stream) {
  (void)in_sizes; (void)n_in; (void)out_size; (void)ws_size;
  const float* pred = (const float*)d_in[0];
  const float* tgt  = (const float*)d_in[1];
  float* out = (float*)d_out;
  float* acc = (float*)d_ws;   // [0..20]=class_sum, [21..41]=n_c

  cb_zero <<<1, 64, 0, stream>>>(acc);
  cb_main <<<NBLK, BLK_THREADS, 0, stream>>>(pred, tgt, acc);
  cb_final<<<1, 32, 0, stream>>>(acc, out);
}